// Model_23880018165868
// MI455X (gfx1250) — compile-verified
//
#include <hip/hip_runtime.h>

typedef __attribute__((ext_vector_type(2))) float v2f;
typedef __attribute__((ext_vector_type(8))) float v8f;

#define Bb 64
#define Nn 8192
#define Ff 32
#define Gg 32
#define G3 (Gg*Gg*Gg)

// ---------------- Kernel 1: per-batch coord mean ----------------
__global__ void __launch_bounds__(256) center_kernel(const float* __restrict__ coords,
                                                     float* __restrict__ center) {
    __shared__ float sx[256], sy[256], sz[256];
    const int b = blockIdx.x;
    const int t = threadIdx.x;
    const float* cb = coords + (size_t)b * Nn * 3;
    float ax = 0.f, ay = 0.f, az = 0.f;
    for (int i = t; i < Nn; i += 256) {
        ax += cb[i * 3 + 0];
        ay += cb[i * 3 + 1];
        az += cb[i * 3 + 2];
    }
    sx[t] = ax; sy[t] = ay; sz[t] = az;
    __syncthreads();
    for (int s = 128; s > 0; s >>= 1) {
        if (t < s) { sx[t] += sx[t + s]; sy[t] += sy[t + s]; sz[t] += sz[t + s]; }
        __syncthreads();
    }
    if (t == 0) {
        center[b * 3 + 0] = sx[0] * (1.0f / Nn);
        center[b * 3 + 1] = sy[0] * (1.0f / Nn);
        center[b * 3 + 2] = sz[0] * (1.0f / Nn);
    }
}

// ---------------- Kernel 2: zero the 256 MB output (b128 stores) ----------------
__global__ void __launch_bounds__(256) zero_kernel(float4* __restrict__ out, int n4) {
    int i = blockIdx.x * blockDim.x + threadIdx.x;
    if (i < n4) out[i] = make_float4(0.f, 0.f, 0.f, 0.f);
}

// ---------------- Kernel 3: fused WMMA linear + voxel scatter-add ----------------
// One wave32 handles 16 atoms: D(16x32) = A(16x32 feats) x W^T(32x32) + bias,
// via V_WMMA_F32_16X16X4_F32 (K in 8 chunks of 4, two N-halves), then
// global_atomic_add_f32 into out[b, n, ix, iy, iz].
__global__ void __launch_bounds__(256) fused_kernel(const float* __restrict__ coords,
                                                    const float* __restrict__ features,
                                                    const float* __restrict__ W,
                                                    const float* __restrict__ bias,
                                                    const float* __restrict__ center,
                                                    float* __restrict__ out) {
    const int lane = threadIdx.x & 31;
    const int wave = threadIdx.x >> 5;
    const int gw   = blockIdx.x * 8 + wave;   // 0 .. 32767
    const int b    = gw >> 9;                 // / (Nn/16) = /512
    const int tile = gw & 511;
    const int base = tile * 16;
    const int l    = lane & 15;               // row / column within tile
    const int h    = lane >> 4;               // half-wave: which K pair

    // ---- voxel index for atom (base + l); lanes 16-31 mirror lanes 0-15 (no divergence)
    const float* cp = coords + ((size_t)b * Nn + base + l) * 3;
    float sxv = cp[0] - center[b * 3 + 0] + 16.0f;
    float syv = cp[1] - center[b * 3 + 1] + 16.0f;
    float szv = cp[2] - center[b * 3 + 2] + 16.0f;
    int ix = min(max((int)sxv, 0), Gg - 1);   // (int) truncates toward zero == astype(int32)
    int iy = min(max((int)syv, 0), Gg - 1);
    int iz = min(max((int)szv, 0), Gg - 1);
    int vox = (ix * Gg + iy) * Gg + iz;

    // ---- WMMA f32 16x16x4, K = 32 in 8 chunks
    const float* frow = features + ((size_t)b * Nn + base + l) * Ff; // A row m=l
    const float* w0   = W + (size_t)l * Ff;          // B col n = l      (W row-major: W[n][k])
    const float* w1   = W + (size_t)(l + 16) * Ff;   // B col n = l + 16

    v8f c0 = {};   // D columns  0..15
    v8f c1 = {};   // D columns 16..31
#pragma unroll
    for (int c = 0; c < 8; ++c) {
        const int k = 4 * c + 2 * h;   // lanes 0-15: K={4c,4c+1}; lanes 16-31: K={4c+2,4c+3}
        v2f a, b0m, b1m;
        a.x   = frow[k]; a.y   = frow[k + 1];
        b0m.x = w0[k];   b0m.y = w0[k + 1];
        b1m.x = w1[k];   b1m.y = w1[k + 1];
        c0 = __builtin_amdgcn_wmma_f32_16x16x4_f32(false, a, false, b0m, (short)0, c0, false, false);
        c1 = __builtin_amdgcn_wmma_f32_16x16x4_f32(false, a, false, b1m, (short)0, c1, false, false);
    }

    // ---- bias + scatter. D layout: VGPR v -> row (lane<16 ? v : v+8), col n0+(lane&15).
    const float bias0 = bias[l];
    const float bias1 = bias[l + 16];
    float* out0 = out + ((size_t)(b * Ff + l)      ) * (size_t)G3;
    float* out1 = out + ((size_t)(b * Ff + l + 16) ) * (size_t)G3;
#pragma unroll
    for (int v = 0; v < 8; ++v) {
        const int m  = h ? (v + 8) : v;          // atom row this lane holds in VGPR v
        const int vm = __shfl(vox, m, 32);       // voxel of atom (base + m)
        atomicAdd(out0 + vm, c0[v] + bias0);
        atomicAdd(out1 + vm, c1[v] + bias1);
    }
}

extern "C" void kernel_launch(void* const* d_in, const int* in_sizes, int n_in,
                              void* d_out, int out_size, void* d_ws, size_t ws_size,
                              hipStream_t stream) {
    const float* coords   = (const float*)d_in[0];  // (B, N, 3)
    const float* features = (const float*)d_in[1];  // (B, N, F)
    const float* W        = (const float*)d_in[2];  // (F, F)
    const float* bias     = (const float*)d_in[3];  // (F,)
    float* out    = (float*)d_out;                  // (B, F, G, G, G)
    float* center = (float*)d_ws;                   // (B, 3) scratch

    center_kernel<<<Bb, 256, 0, stream>>>(coords, center);

    const int n4 = out_size / 4;                    // 16,777,216 float4
    zero_kernel<<<(n4 + 255) / 256, 256, 0, stream>>>((float4*)out, n4);

    // 64 batches * 512 tiles = 32768 waves; 8 waves / block
    fused_kernel<<<(Bb * (Nn / 16)) / 8, 256, 0, stream>>>(coords, features, W, bias, center, out);
}